// GruBlock_v_4492535792147
// MI455X (gfx1250) — compile-verified
//
#include <hip/hip_runtime.h>

typedef __attribute__((ext_vector_type(2))) float v2f;
typedef __attribute__((ext_vector_type(8))) float v8f;

#define B_N   64
#define C_IN  64
#define H_IMG 48
#define W_IMG 160
#define NSEQ  (B_N * H_IMG)          // 3072
#define GW    192                    // 2 * 3 * H_GRU
#define HG    32

__device__ __forceinline__ float sigf(float x) { return 1.0f / (1.0f + expf(-x)); }

// ---------------------------------------------------------------------------
// Kernel 1: fuse conv (1x1) into the GRU input projection.
//   G[g][c]  = sum_o wih[g][o] * conv_w[o][c]        (g<96: fwd, g>=96: bwd)
//   Gb[g]    = sum_o wih[g][o] * conv_b[o] + bih[g]
// ---------------------------------------------------------------------------
__global__ __launch_bounds__(64) void fuse_weights(
    const float* __restrict__ conv_w, const float* __restrict__ conv_b,
    const float* __restrict__ wih_f, const float* __restrict__ bih_f,
    const float* __restrict__ wih_b, const float* __restrict__ bih_b,
    float* __restrict__ G, float* __restrict__ Gb) {
  const int g = blockIdx.x;      // 0..191
  const int c = threadIdx.x;     // 0..63
  const float* wih = (g < 96) ? (wih_f + g * 64) : (wih_b + (g - 96) * 64);
  float s = 0.0f;
#pragma unroll 8
  for (int o = 0; o < 64; ++o) s += wih[o] * conv_w[o * 64 + c];
  G[g * 64 + c] = s;
  if (c == 0) {
    float sb = 0.0f;
#pragma unroll 8
    for (int o = 0; o < 64; ++o) sb += wih[o] * conv_b[o];
    Gb[g] = sb + ((g < 96) ? bih_f[g] : bih_b[g - 96]);
  }
}

// ---------------------------------------------------------------------------
// Kernel 2: pre-projection GEMM (491520 x 64) @ (64 x 192) via WMMA f32.
// One wave per 16-pixel M-tile (tiles never straddle a sequence: 160 % 16 == 0).
// pre[pixel][g] = sum_c x_pixel[c] * G[g][c] + Gb[g]
// ---------------------------------------------------------------------------
__global__ __launch_bounds__(256) void pre_gemm(
    const float* __restrict__ x, const float* __restrict__ G,
    const float* __restrict__ Gb, float* __restrict__ pre) {
  const int lane = threadIdx.x & 31;
  const int wv   = threadIdx.x >> 5;           // 0..7
  const int mt   = blockIdx.x * 8 + wv;        // 0..30719
  const int p0   = mt * 16;                    // first pixel of tile
  const int seq  = p0 / W_IMG;
  const int w0   = p0 % W_IMG;
  const int b    = seq / H_IMG;
  const int hh   = seq % H_IMG;
  const int lo = lane & 15, hi = lane >> 4;

  // x[b][c][hh][w0+lo], channel stride = H*W = 7680
  const long xbase = (long)b * (C_IN * H_IMG * W_IMG) + (long)hh * W_IMG + w0 + lo;

  v2f afrag[16];
#pragma unroll
  for (int kc = 0; kc < 16; ++kc) {
    const int c0 = kc * 4 + hi * 2;
    afrag[kc].x = x[xbase + (long)c0 * (H_IMG * W_IMG)];
    afrag[kc].y = x[xbase + (long)(c0 + 1) * (H_IMG * W_IMG)];
  }

#pragma unroll
  for (int n0 = 0; n0 < 12; ++n0) {
    v8f acc = {};
    const int grow = n0 * 16 + lo;
#pragma unroll
    for (int kc = 0; kc < 16; ++kc) {
      const int c0 = kc * 4 + hi * 2;
      v2f bf;
      bf.x = G[grow * 64 + c0];
      bf.y = G[grow * 64 + c0 + 1];
      acc = __builtin_amdgcn_wmma_f32_16x16x4_f32(
          false, afrag[kc], false, bf, (short)0, acc, false, false);
    }
    const float bias = Gb[grow];
#pragma unroll
    for (int r = 0; r < 8; ++r) {
      const int m = hi * 8 + r;                 // row within tile
      pre[(long)(p0 + m) * GW + n0 * 16 + lo] = acc[r] + bias;
    }
  }
}

// ---------------------------------------------------------------------------
// Kernel 3: bidirectional GRU scan. One wave handles 16 sequences of one
// direction. Per step: gh(16x96) = h(16x32) @ whh^T via 6x8 WMMA f32
// 16x16x4, gate math in D-layout, LDS transpose back to A-fragments.
// ---------------------------------------------------------------------------
__global__ __launch_bounds__(32) void gru_scan(
    const float* __restrict__ pre,
    const float* __restrict__ whh_f, const float* __restrict__ bhh_f,
    const float* __restrict__ whh_b, const float* __restrict__ bhh_b,
    float* __restrict__ out) {
  __shared__ float hs[16][33];                 // [seq][unit], padded
  const int lane = threadIdx.x;
  const int lo = lane & 15, hi = lane >> 4;
  const int dir  = blockIdx.x / (NSEQ / 16);   // 0 = fwd, 1 = bwd
  const int blk  = blockIdx.x % (NSEQ / 16);
  const int seq0 = blk * 16;
  const float* whh = dir ? whh_b : whh_f;
  const float* bhh = dir ? bhh_b : bhh_f;

  // Loop-invariant recurrent B-fragments: B[k][n] = whh[n0*16+n][k]
  v2f bfrag[6][8];
  float bh[6];
#pragma unroll
  for (int n0 = 0; n0 < 6; ++n0) {
    const int grow = n0 * 16 + lo;
#pragma unroll
    for (int kc = 0; kc < 8; ++kc) {
      const int k0 = kc * 4 + hi * 2;
      bfrag[n0][kc].x = whh[grow * HG + k0];
      bfrag[n0][kc].y = whh[grow * HG + k0 + 1];
    }
    bh[n0] = bhh[grow];
  }

  // Per-(p,r) output bases: out[b][dir*32+u][hh][w], u = p*16+lo
  long obase[2][8];
  long pbase[8];
#pragma unroll
  for (int r = 0; r < 8; ++r) {
    const int seq = seq0 + hi * 8 + r;
    const int b = seq / H_IMG, hh2 = seq % H_IMG;
    const long base = ((long)(b * 64 + dir * HG) * H_IMG + hh2) * W_IMG;
    obase[0][r] = base + (long)lo * (H_IMG * W_IMG);
    obase[1][r] = base + (long)(lo + 16) * (H_IMG * W_IMG);
    pbase[r] = (long)seq * W_IMG * GW + dir * 96 + lo;
  }

  v2f afrag[8];
#pragma unroll
  for (int kc = 0; kc < 8; ++kc) { afrag[kc].x = 0.0f; afrag[kc].y = 0.0f; }
  v8f hD[2];
  hD[0] = {}; hD[1] = {};

  for (int step = 0; step < W_IMG; ++step) {
    const int w = dir ? (W_IMG - 1 - step) : step;

    // gi in D-layout: gi[n0][r] = pre[seq(hi,r)][w][dir*96 + n0*16 + lo]
    v8f gi[6];
#pragma unroll
    for (int n0 = 0; n0 < 6; ++n0)
#pragma unroll
      for (int r = 0; r < 8; ++r)
        gi[n0][r] = pre[pbase[r] + (long)w * GW + n0 * 16];

    // gh = h @ whh^T   (6 output tiles, K = 32 -> 8 chained WMMAs each)
    v8f acc[6];
#pragma unroll
    for (int n0 = 0; n0 < 6; ++n0) {
      v8f a = {};
#pragma unroll
      for (int kc = 0; kc < 8; ++kc)
        a = __builtin_amdgcn_wmma_f32_16x16x4_f32(
            false, afrag[kc], false, bfrag[n0][kc], (short)0, a, false, false);
      acc[n0] = a;
    }

    // Gates + state update in D-layout; emit output; stage transpose in LDS.
#pragma unroll
    for (int p = 0; p < 2; ++p) {
#pragma unroll
      for (int r = 0; r < 8; ++r) {
        const float rg = sigf(gi[0 + p][r] + acc[0 + p][r] + bh[0 + p]);
        const float zg = sigf(gi[2 + p][r] + acc[2 + p][r] + bh[2 + p]);
        const float ng = tanhf(gi[4 + p][r] + rg * (acc[4 + p][r] + bh[4 + p]));
        const float hn = (1.0f - zg) * ng + zg * hD[p][r];
        hD[p][r] = hn;
        hs[hi * 8 + r][p * 16 + lo] = hn;
        out[obase[p][r] + w] = hn;
      }
    }
    __syncthreads();   // single-wave WG: NOP barrier, DScnt waits inserted
#pragma unroll
    for (int kc = 0; kc < 8; ++kc) {
      const int k0 = kc * 4 + hi * 2;
      afrag[kc].x = hs[lo][k0];
      afrag[kc].y = hs[lo][k0 + 1];
    }
    __syncthreads();
  }
}

// ---------------------------------------------------------------------------
extern "C" void kernel_launch(void* const* d_in, const int* in_sizes, int n_in,
                              void* d_out, int out_size, void* d_ws, size_t ws_size,
                              hipStream_t stream) {
  const float* x      = (const float*)d_in[0];
  const float* conv_w = (const float*)d_in[1];
  const float* conv_b = (const float*)d_in[2];
  const float* wih_f  = (const float*)d_in[3];
  const float* whh_f  = (const float*)d_in[4];
  const float* bih_f  = (const float*)d_in[5];
  const float* bhh_f  = (const float*)d_in[6];
  const float* wih_b  = (const float*)d_in[7];
  const float* whh_b  = (const float*)d_in[8];
  const float* bih_b  = (const float*)d_in[9];
  const float* bhh_b  = (const float*)d_in[10];
  float* out = (float*)d_out;

  float* G   = (float*)d_ws;            // 192*64 floats
  float* Gb  = G + GW * 64;             // 192 floats
  float* pre = (float*)d_ws + 16384;    // 3072*160*192 floats (~377 MB)

  fuse_weights<<<GW, 64, 0, stream>>>(conv_w, conv_b, wih_f, bih_f, wih_b, bih_b, G, Gb);

  const int n_mtiles = (NSEQ * W_IMG) / 16;          // 30720
  pre_gemm<<<n_mtiles / 8, 256, 0, stream>>>(x, G, Gb, pre);

  gru_scan<<<2 * (NSEQ / 16), 32, 0, stream>>>(pre, whh_f, bhh_f, whh_b, bhh_b, out);
}